// GCN_67920612819568
// MI455X (gfx1250) — compile-verified
//
#include <hip/hip_runtime.h>

// ---------------------------------------------------------------------------
// GCN forward on MI455X (gfx1250, wave32).
// Dense GEMMs use V_WMMA_F32_16X16X4_F32 (f32 in / f32 acc, matches the f32
// reference). B is staged TRANSPOSED in LDS (Blds[n][k], k-padded by 2) so
// each B fragment is a single conflict-free ds_load_b64; each wave computes
// TWO 16x16 M-tiles so every B fragment feeds two back-to-back WMMAs.
// SpMM is scatter-add via global f32 atomics; t/agg (25.6MB each) fit in the
// 192MB L2 so the random edge traffic stays on-chip.
// ---------------------------------------------------------------------------

typedef float v2f __attribute__((ext_vector_type(2)));
typedef float v8f __attribute__((ext_vector_type(8)));

#define F_IN      128
#define F_H       64
#define N_CLS     40
#define N_CLS_PAD 48
#define KPAD      2          // LDS k-padding: bank-conflict-free b64 loads

// ---------------------------------------------------------------------------
// WMMA GEMM: C[M x N] = A[M x K] @ B[K x N] (+ bias).
// blockDim = 32 * (Npad/16); grid.x = M/32 (M = 100000 = 3125 * 32).
// Wave w computes rows [32*blk, 32*blk+32) x cols [16*w, 16*w+16).
// ---------------------------------------------------------------------------
__global__ void wmma_gemm_kernel(const float* __restrict__ A,
                                 const float* __restrict__ B,
                                 const float* __restrict__ bias,
                                 float* __restrict__ C,
                                 int K, int N, int Npad)
{
    __shared__ float Blds[64 * (F_IN + KPAD)];   // max 64 x 130 f32 = 33.3KB
    const int KP = K + KPAD;

    // Cooperative transpose-load of B into LDS: Blds[n*KP + k] = B[k*N + n],
    // zero-padding cols N..Npad and k-rows K..KP.
    for (int i = threadIdx.x; i < Npad * KP; i += blockDim.x) {
        int n = i / KP;
        int k = i - n * KP;
        Blds[i] = (k < K && n < N) ? B[k * N + n] : 0.0f;
    }
    __syncthreads();

    const int lane    = threadIdx.x & 31;
    const int wave    = threadIdx.x >> 5;
    const int rowBase = blockIdx.x << 5;       // 32 rows per block
    const int colBase = wave << 4;

    // A fragment (16x4 f32): lanes 0-15 -> row, K={k0,k0+1};
    //                        lanes 16-31 -> row, K={k0+2,k0+3}.
    const int khalf = (lane >> 4) << 1;        // 0 or 2
    const int mrow0 = rowBase + (lane & 15);
    const int mrow1 = mrow0 + 16;
    // B fragment (4x16 f32): lane&15 gives N index; same K-half split.
    const int ncol  = colBase + (lane & 15);

    const float* Arow0 = A + (long long)mrow0 * K + khalf;
    const float* Arow1 = A + (long long)mrow1 * K + khalf;
    const float* Bcol  = Blds + ncol * KP + khalf;

    v8f c0 = {};
    v8f c1 = {};
    for (int k0 = 0; k0 < K; k0 += 4) {
        v2f a0 = *reinterpret_cast<const v2f*>(Arow0 + k0);  // global_load_b64
        v2f a1 = *reinterpret_cast<const v2f*>(Arow1 + k0);  // global_load_b64
        v2f b  = *reinterpret_cast<const v2f*>(Bcol + k0);   // ds_load_b64
        // D = A*B + C  (neg_a, A, neg_b, B, c_mod, C, reuse_a, reuse_b)
        c0 = __builtin_amdgcn_wmma_f32_16x16x4_f32(
                 false, a0, false, b, (short)0, c0, false, false);
        c1 = __builtin_amdgcn_wmma_f32_16x16x4_f32(
                 false, a1, false, b, (short)0, c1, false, false);
    }

    // D layout: VGPR j, lanes 0-15 -> M=j, lanes 16-31 -> M=j+8; N = lane&15.
    const int mOff = (lane >> 4) << 3;
    const int col  = colBase + (lane & 15);
    float bv = 0.0f;
    if (bias != nullptr && col < N) bv = bias[col];
    if (col < N) {
#pragma unroll
        for (int j = 0; j < 8; ++j) {
            int row = rowBase + j + mOff;
            C[(long long)row * N + col]        = c0[j] + bv;
            C[(long long)(row + 16) * N + col] = c1[j] + bv;
        }
    }
}

// ---------------------------------------------------------------------------
// SpMM scatter: agg[dst[e], :] += w[e] * T[src[e], :]
// 16 threads per edge, each moving a float4 chunk of the 64-wide feature row.
// Atomics are no-return f32 adds (STOREcnt path); agg lives in L2.
// ---------------------------------------------------------------------------
__global__ void spmm_kernel(const int* __restrict__ esrc,
                            const int* __restrict__ edst,
                            const float* __restrict__ ew,
                            const float* __restrict__ T,
                            float* __restrict__ agg,
                            long long total)
{
    long long t = (long long)blockIdx.x * blockDim.x + threadIdx.x;
    if (t >= total) return;
    int e  = (int)(t >> 4);
    int c4 = (int)(t & 15) << 2;
    int s  = esrc[e];
    int d  = edst[e];
    float w = ew[e];
    const float4 v = *reinterpret_cast<const float4*>(T + (long long)s * F_H + c4);
    float* o = agg + (long long)d * F_H + c4;
    atomicAdd(o + 0, v.x * w);
    atomicAdd(o + 1, v.y * w);
    atomicAdd(o + 2, v.z * w);
    atomicAdd(o + 3, v.w * w);
}

// ---------------------------------------------------------------------------
// h = relu(agg + b)   (feature dim F_H = 64, power of two)
// ---------------------------------------------------------------------------
__global__ void bias_relu_kernel(const float* __restrict__ agg,
                                 const float* __restrict__ b,
                                 float* __restrict__ h,
                                 int total)
{
    int t = blockIdx.x * blockDim.x + threadIdx.x;
    if (t >= total) return;
    float v = agg[t] + b[t & (F_H - 1)];
    h[t] = v > 0.0f ? v : 0.0f;
}

// ---------------------------------------------------------------------------
// Row-wise log-softmax over N_CLS = 40 columns; one thread per row.
// ---------------------------------------------------------------------------
__global__ void log_softmax_kernel(const float* __restrict__ logits,
                                   float* __restrict__ logp,
                                   int nrows)
{
    int r = blockIdx.x * blockDim.x + threadIdx.x;
    if (r >= nrows) return;
    const float* x = logits + (long long)r * N_CLS;
    float m = x[0];
#pragma unroll
    for (int i = 1; i < N_CLS; ++i) m = fmaxf(m, x[i]);
    float s = 0.0f;
#pragma unroll
    for (int i = 0; i < N_CLS; ++i) s += expf(x[i] - m);
    float ls = logf(s) + m;
    float* o = logp + (long long)r * N_CLS;
#pragma unroll
    for (int i = 0; i < N_CLS; ++i) o[i] = x[i] - ls;
}

// ---------------------------------------------------------------------------
// Host launcher
// ---------------------------------------------------------------------------
extern "C" void kernel_launch(void* const* d_in, const int* in_sizes, int n_in,
                              void* d_out, int out_size, void* d_ws, size_t ws_size,
                              hipStream_t stream)
{
    const float* x    = (const float*)d_in[0];
    const int*   esrc = (const int*)  d_in[1];
    const int*   edst = (const int*)  d_in[2];
    const float* ew   = (const float*)d_in[3];
    const float* W1   = (const float*)d_in[4];
    const float* b1   = (const float*)d_in[5];
    const float* W2   = (const float*)d_in[6];
    const float* b2   = (const float*)d_in[7];
    const float* Wl   = (const float*)d_in[8];
    const float* bl   = (const float*)d_in[9];

    const int N = in_sizes[0] / F_IN;   // 100000 nodes
    const int E = in_sizes[1];          // 1.6M edges

    // Output layout: (logp[N,40], h1[N,64], h2[N,64], logits[N,40]) flat.
    float* out    = (float*)d_out;
    float* logp   = out;
    float* h1     = logp + (long long)N * N_CLS;
    float* h2     = h1   + (long long)N * F_H;
    float* logits = h2   + (long long)N * F_H;

    // Workspace: t (N x 64) reused by both layers, agg (N x 64) accumulator.
    float* t   = (float*)d_ws;
    float* agg = t + (long long)N * F_H;

    const int       gemmBlocks = (N + 31) / 32;            // 3125
    const long long totalE     = (long long)E * (F_H / 4); // E * 16 chunks
    const int       spmmBlocks = (int)((totalE + 255) / 256);
    const int       ewBlocks   = (N * F_H + 255) / 256;

    // ---- Layer 1: h1 = relu(spmm(x @ W1) + b1) ----
    wmma_gemm_kernel<<<gemmBlocks, 128, 0, stream>>>(x, W1, nullptr, t,
                                                     F_IN, F_H, F_H);
    hipMemsetAsync(agg, 0, (size_t)N * F_H * sizeof(float), stream);
    spmm_kernel<<<spmmBlocks, 256, 0, stream>>>(esrc, edst, ew, t, agg, totalE);
    bias_relu_kernel<<<ewBlocks, 256, 0, stream>>>(agg, b1, h1, N * F_H);

    // ---- Layer 2: h2 = relu(spmm(h1 @ W2) + b2) ----
    wmma_gemm_kernel<<<gemmBlocks, 128, 0, stream>>>(h1, W2, nullptr, t,
                                                     F_H, F_H, F_H);
    hipMemsetAsync(agg, 0, (size_t)N * F_H * sizeof(float), stream);
    spmm_kernel<<<spmmBlocks, 256, 0, stream>>>(esrc, edst, ew, t, agg, totalE);
    bias_relu_kernel<<<ewBlocks, 256, 0, stream>>>(agg, b2, h2, N * F_H);

    // ---- Head: logits = h2 @ Wl + bl ; logp = log_softmax(logits) ----
    wmma_gemm_kernel<<<gemmBlocks, 96, 0, stream>>>(h2, Wl, bl, logits,
                                                    F_H, N_CLS, N_CLS_PAD);
    log_softmax_kernel<<<(N + 255) / 256, 256, 0, stream>>>(logits, logp, N);
}